// Attention_1554778161179
// MI455X (gfx1250) — compile-verified
//
#include <hip/hip_runtime.h>
#include <math.h>

typedef float v2f __attribute__((ext_vector_type(2)));
typedef float v8f __attribute__((ext_vector_type(8)));

#define BATCH 32
#define QL 16
#define KVC 8192
#define DIM 128
#define CHUNK 512
#define NCHC (KVC / CHUNK)   /* 16 cache chunks */
#define NCH (NCHC + 1)       /* + 1 new-token chunk */
#define KSTRIDE 132          /* padded LDS row stride (words): bank-conflict-free */
#define SCALE 0.08838834764831845f

static __device__ __forceinline__ v8f wmma4(v2f a, v2f b, v8f c) {
  // 8 args: (neg_a, A, neg_b, B, c_mod, C, reuse_a, reuse_b)
  return __builtin_amdgcn_wmma_f32_16x16x4_f32(false, a, false, b, (short)0, c,
                                               false, false);
}

// Async copy 16 B/lane from global to LDS (GLOBAL_LOAD_ASYNC_TO_LDS_B128,
// tracked by ASYNCcnt).  LDS offset = low 32 bits of the generic pointer
// (flat->LDS aperture truncation rule, ISA 10.2).
static __device__ __forceinline__ void async_copy16(const float* gsrc,
                                                    const float* lds_dst) {
  unsigned loff = (unsigned)(__SIZE_TYPE__)lds_dst;
  unsigned long long gaddr = (unsigned long long)(__SIZE_TYPE__)gsrc;
  asm volatile("global_load_async_to_lds_b128 %0, %1, off"
               :: "v"(loff), "v"(gaddr)
               : "memory");
}

static __device__ __forceinline__ void wait_async_0() {
  asm volatile("s_wait_asynccnt 0" ::: "memory");
}
static __device__ __forceinline__ void wait_async_16() {
  asm volatile("s_wait_asynccnt 16" ::: "memory");
}

// Stage one 16x128 f32 K tile into LDS (padded rows). 16 async instructions,
// each moving one 512B row fully coalesced.
static __device__ __forceinline__ void stage_k_tile(const float* gbase,
                                                    const float* lbase,
                                                    int lane) {
#pragma unroll
  for (int r = 0; r < 16; ++r)
    async_copy16(gbase + r * DIM + lane * 4, lbase + r * KSTRIDE + lane * 4);
}

// ---------------------------------------------------------------------------
// Per-tile flash-attention step (computes S^T = K @ Q^T so the softmax
// reduction over keys is in-lane + one xor-16 exchange).
// kt/lp/la are references to named __shared__ arrays -> ds_* ops guaranteed.
// ---------------------------------------------------------------------------
static __device__ __forceinline__ void compute_tile(
    const float (&kt)[16 * KSTRIDE], float (&lp)[16 * 16], float (&la)[16],
    const float* __restrict__ Vp, int kbase, const v2f (&qa)[32],
    float& m, float& l, v8f (&acc)[8], int n, int h) {
  // S^T tile [16k x 16q] = K_tile @ Q^T.  A-operand (K) from LDS,
  // B-operand (Q^T) preloaded in registers.  Fully unrolled so qa stays
  // in VGPRs (no dynamic indexing -> no scratch demotion).
  v8f s = {};
#pragma unroll
  for (int j = 0; j < 32; ++j) {
    int kk = 4 * j + 2 * h;
    v2f ka = { kt[n * KSTRIDE + kk], kt[n * KSTRIDE + kk + 1] };
    s = wmma4(ka, qa[j], s);
  }

  // Per-query (= per-lane) online softmax over this tile's 16 keys:
  // keys r+8h live in-lane (8 values), the other 8 in the partner half.
  float sv[8];
#pragma unroll
  for (int r = 0; r < 8; ++r) sv[r] = s[r] * SCALE;
  float tm = sv[0];
#pragma unroll
  for (int r = 1; r < 8; ++r) tm = fmaxf(tm, sv[r]);
  tm = fmaxf(tm, __shfl_xor(tm, 16, 32));   // single cross-half exchange
  float mn = fmaxf(m, tm);
  float al = __expf(m - mn);
  m = mn;
  float p8[8];
  float rs = 0.0f;
#pragma unroll
  for (int r = 0; r < 8; ++r) {
    p8[r] = __expf(sv[r] - mn);
    rs += p8[r];
  }
  rs += __shfl_xor(rs, 16, 32);             // single cross-half exchange
  l = l * al + rs;

  // Relay P^T tile (key-major) and per-query alpha through LDS.
#pragma unroll
  for (int r = 0; r < 8; ++r) lp[(r + 8 * h) * 16 + n] = p8[r];
  la[n] = al;                               // both halves write same value
  __syncthreads();
  float alar[8];
#pragma unroll
  for (int r = 0; r < 8; ++r) alar[r] = la[r + 8 * h];  // broadcast reads
  v2f pa[4];
#pragma unroll
  for (int j = 0; j < 4; ++j) {
    int kk = 4 * j + 2 * h;
    pa[j] = { lp[kk * 16 + n], lp[(kk + 1) * 16 + n] };
  }
  __syncthreads();

  // Rescale accumulators by alpha(row q = r + 8h).
#pragma unroll
  for (int td = 0; td < 8; ++td)
#pragma unroll
    for (int r = 0; r < 8; ++r) acc[td][r] *= alar[r];

  // acc[16q x 128d] += P[16x16] @ V[16k x 128d]  (V direct: coalesced)
#pragma unroll
  for (int td = 0; td < 8; ++td) {
    int d0 = td * 16;
#pragma unroll
    for (int j = 0; j < 4; ++j) {
      int kk = 4 * j + 2 * h;
      v2f bb = { Vp[(size_t)(kbase + kk) * DIM + d0 + n],
                 Vp[(size_t)(kbase + kk + 1) * DIM + d0 + n] };
      acc[td] = wmma4(pa[j], bb, acc[td]);
    }
  }
}

// ---------------------------------------------------------------------------
// Kernel: flash-attention partial over one KV chunk per single-wave block.
// K tiles double-buffered in LDS via global_load_async_to_lds_b128.
// ---------------------------------------------------------------------------
__global__ void __launch_bounds__(32)
flash_partial(const float* __restrict__ cacheK,
              const float* __restrict__ cacheV,
              const float* __restrict__ Qp,
              const float* __restrict__ Kn,
              const float* __restrict__ Vn,
              float* __restrict__ Wm, float* __restrict__ Wl,
              float* __restrict__ Wacc) {
  __shared__ __align__(16) float kb0[16 * KSTRIDE];   // K tile ping
  __shared__ __align__(16) float kb1[16 * KSTRIDE];   // K tile pong
  __shared__ __align__(16) float lp[16 * 16];         // P-tile relay
  __shared__ __align__(16) float la[16];              // alpha relay

  int b  = blockIdx.x / NCH;
  int ch = blockIdx.x % NCH;
  int lane = threadIdx.x;
  int n = lane & 15;
  int h = lane >> 4;

  const float* Kp;
  const float* Vp;
  int ntile;
  if (ch < NCHC) {
    size_t off = ((size_t)b * KVC + (size_t)ch * CHUNK) * DIM;
    Kp = cacheK + off;
    Vp = cacheV + off;
    ntile = CHUNK / 16;
  } else {
    Kp = Kn + b * (QL * DIM);
    Vp = Vn + b * (QL * DIM);
    ntile = 1;
  }

  // Preload Q (16x128) as 32 B-layout slices — reused every tile.
  const float* Qb = Qp + b * (QL * DIM);
  v2f qa[32];
#pragma unroll
  for (int j = 0; j < 32; ++j) {
    int kk = 4 * j + 2 * h;
    qa[j] = { Qb[n * DIM + kk], Qb[n * DIM + kk + 1] };
  }

  float m = -INFINITY, l = 0.0f;
  v8f acc[8];
#pragma unroll
  for (int t = 0; t < 8; ++t) acc[t] = {};

  // Prologue: stage K tile 0.
  stage_k_tile(Kp, kb0, lane);

  for (int t = 0; t < ntile; ++t) {
    int kbase = t * 16;

    if (t + 1 < ntile) {
      // Prefetch next K tile into the other buffer; wait for current tile.
      const float* nb = ((t + 1) & 1) ? kb1 : kb0;
      stage_k_tile(Kp + (size_t)(t + 1) * 16 * DIM, nb, lane);
      wait_async_16();
      // Prefetch next V tile (8KB = 2 cachelines per lane).
      const float* vn0 = Vp + (size_t)(kbase + 16) * DIM;
      __builtin_prefetch(vn0 + lane * 64, 0, 3);
      __builtin_prefetch(vn0 + lane * 64 + 32, 0, 3);
    } else {
      wait_async_0();
    }

    if ((t & 1) == 0)
      compute_tile(kb0, lp, la, Vp, kbase, qa, m, l, acc, n, h);
    else
      compute_tile(kb1, lp, la, Vp, kbase, qa, m, l, acc, n, h);
  }

  // Write partials (m, l, acc) for this (batch, chunk).
  int pi = b * NCH + ch;
  if (lane < 16) {
    Wm[pi * QL + lane] = m;
    Wl[pi * QL + lane] = l;
  }
#pragma unroll
  for (int td = 0; td < 8; ++td)
#pragma unroll
    for (int r = 0; r < 8; ++r) {
      int q = r + 8 * h;
      Wacc[((size_t)pi * QL + q) * DIM + td * 16 + n] = acc[td][r];
    }
}

// ---------------------------------------------------------------------------
// Kernel: Q/K/V projections.  One wave per (batch, proj, 16-col tile).
// ---------------------------------------------------------------------------
__global__ void __launch_bounds__(32)
proj_kernel(const float* __restrict__ X,
            const float* __restrict__ Wq, const float* __restrict__ bq,
            const float* __restrict__ Wk, const float* __restrict__ bk,
            const float* __restrict__ Wv, const float* __restrict__ bv,
            float* __restrict__ Qp, float* __restrict__ Kn,
            float* __restrict__ Vn) {
  int idx = blockIdx.x;
  int b = idx / 24;
  int rem = idx % 24;
  int p = rem >> 3;              // 0=Q 1=K 2=V
  int n0 = (rem & 7) * 16;       // output-column tile

  const float* W    = (p == 0) ? Wq : (p == 1) ? Wk : Wv;
  const float* bias = (p == 0) ? bq : (p == 1) ? bk : bv;
  float* out = ((p == 0) ? Qp : (p == 1) ? Kn : Vn) + b * (QL * DIM);
  const float* Xb = X + b * (QL * DIM);

  int lane = threadIdx.x;
  int n = lane & 15;
  int h = lane >> 4;

  v8f c = {};
#pragma unroll 8
  for (int j = 0; j < 32; ++j) {
    int kk = 4 * j + 2 * h;
    v2f a  = { Xb[n * DIM + kk], Xb[n * DIM + kk + 1] };
    v2f bb = { W[(n0 + n) * DIM + kk], W[(n0 + n) * DIM + kk + 1] };
    c = wmma4(a, bb, c);
  }
  float bval = bias[n0 + n];
#pragma unroll
  for (int r = 0; r < 8; ++r) {
    int q = r + 8 * h;
    out[q * DIM + n0 + n] = c[r] + bval;
  }
}

// ---------------------------------------------------------------------------
// Kernel: merge the NCH partials per batch (log-sum-exp combine).
// ---------------------------------------------------------------------------
__global__ void __launch_bounds__(256)
reduce_kernel(const float* __restrict__ Wm,
              const float* __restrict__ Wl,
              const float* __restrict__ Wacc,
              float* __restrict__ out) {
  int b = blockIdx.x;
  for (int e = threadIdx.x; e < QL * DIM; e += blockDim.x) {
    int q = e >> 7;
    int d = e & 127;
    float M = -INFINITY;
    for (int c = 0; c < NCH; ++c)
      M = fmaxf(M, Wm[(b * NCH + c) * QL + q]);
    float L = 0.0f, S = 0.0f;
    for (int c = 0; c < NCH; ++c) {
      float w = __expf(Wm[(b * NCH + c) * QL + q] - M);
      L += Wl[(b * NCH + c) * QL + q] * w;
      S += Wacc[((size_t)(b * NCH + c) * QL + q) * DIM + d] * w;
    }
    out[((size_t)b * QL + q) * DIM + d] = S / L;
  }
}

// ---------------------------------------------------------------------------
extern "C" void kernel_launch(void* const* d_in, const int* in_sizes, int n_in,
                              void* d_out, int out_size, void* d_ws,
                              size_t ws_size, hipStream_t stream) {
  const float* X  = (const float*)d_in[0];
  const float* cK = (const float*)d_in[1];
  const float* cV = (const float*)d_in[2];
  const float* Wq = (const float*)d_in[3];
  const float* bq = (const float*)d_in[4];
  const float* Wk = (const float*)d_in[5];
  const float* bk = (const float*)d_in[6];
  const float* Wv = (const float*)d_in[7];
  const float* bv = (const float*)d_in[8];
  float* out = (float*)d_out;
  float* ws  = (float*)d_ws;

  float* Qp   = ws;
  float* Kn   = Qp + BATCH * QL * DIM;
  float* Vn   = Kn + BATCH * QL * DIM;
  float* Wm   = Vn + BATCH * QL * DIM;
  float* Wl   = Wm + BATCH * NCH * QL;
  float* Wacc = Wl + BATCH * NCH * QL;   // 32*17*16*128 floats (~4.4 MB)

  proj_kernel<<<BATCH * 24, 32, 0, stream>>>(X, Wq, bq, Wk, bk, Wv, bv,
                                             Qp, Kn, Vn);
  flash_partial<<<BATCH * NCH, 32, 0, stream>>>(cK, cV, Qp, Kn, Vn,
                                                Wm, Wl, Wacc);
  reduce_kernel<<<BATCH, 256, 0, stream>>>(Wm, Wl, Wacc, out);
}